// shuffer_v2_72773925864321
// MI455X (gfx1250) — compile-verified
//
#include <hip/hip_runtime.h>
#include <hip/hip_bf16.h>

typedef __attribute__((ext_vector_type(2))) float v2f;
typedef __attribute__((ext_vector_type(8))) float v8f;

// ---------------------------------------------------------------------------
// Kernel 1: global-average-pool per (batch, channel) using V_WMMA_F32_16X16X4_F32
// as an exact f32 row-summing engine. One wave32 per channel.
//   A (16x4): lane m (m<16) holds x[base + m*4 + 0..1] (K0,K1),
//             lane m+16 holds x[base + m*4 + 2..3]     (K2,K3)
//   B (4x16): all ones -> D[m][n] = sum_k A[m,k]  (replicated across columns)
//   After the loop: lanes 0..15 hold row sums 0..7 in acc[0..7],
//                   lanes 16..31 hold row sums 8..15. Sum + shfl_xor(16).
// Each WMMA consumes 64 consecutive floats (256B, fully coalesced per wave).
// ---------------------------------------------------------------------------
__global__ void gap_wmma_kernel(const float* __restrict__ x,
                                float* __restrict__ means,
                                int HW) {
  const int wave = threadIdx.x >> 5;
  const int lane = threadIdx.x & 31;
  const int gw = blockIdx.x * 8 + wave;       // flat (b*C + c) channel id
  const int m = lane & 15;
  const int half = lane >> 4;

  const float* p = x + (size_t)gw * HW + m * 4 + half * 2;

  v8f acc = {0.f, 0.f, 0.f, 0.f, 0.f, 0.f, 0.f, 0.f};
  v2f ones;
  ones[0] = 1.0f;
  ones[1] = 1.0f;

  const int iters = HW >> 6;                  // 64 elements per iteration
  for (int t = 0; t < iters; ++t) {
    __builtin_prefetch(p + 2048, 0, 1);       // global_prefetch_b8, ~8KB ahead
    float2 v = *reinterpret_cast<const float2*>(p);
    v2f a;
    a[0] = v.x;
    a[1] = v.y;
    acc = __builtin_amdgcn_wmma_f32_16x16x4_f32(
        /*neg_a=*/false, a, /*neg_b=*/false, ones,
        /*c_mod=*/(short)0, acc, /*reuse_a=*/false, /*reuse_b=*/false);
    p += 64;
  }

  float s = acc[0] + acc[1] + acc[2] + acc[3] + acc[4] + acc[5] + acc[6] + acc[7];
  s += __shfl_xor(s, 16, 32);                 // rows 0..7 + rows 8..15
  if (lane == 0) means[gw] = s / (float)HW;
}

// ---------------------------------------------------------------------------
// Kernel 2: stable ascending argsort per batch via O(C^2) ranking.
// rank(c) = #{k : m_k < m_c  or (m_k == m_c and k < c)}; sidx[rank] = c.
// Matches jnp.argsort (stable, ascending). C <= 384.
// ---------------------------------------------------------------------------
__global__ void rank_argsort_kernel(const float* __restrict__ means,
                                    int* __restrict__ sidx, int C) {
  __shared__ float sm[384];
  const int b = blockIdx.x;
  for (int c = threadIdx.x; c < C; c += blockDim.x) sm[c] = means[b * C + c];
  __syncthreads();
  for (int c = threadIdx.x; c < C; c += blockDim.x) {
    const float mc = sm[c];
    int r = 0;
    for (int k = 0; k < C; ++k) {
      const float mk = sm[k];
      r += (mk < mc) || (mk == mc && k < c);
    }
    sidx[b * C + r] = c;
  }
}

// ---------------------------------------------------------------------------
// Kernel 3: channel-gather + bilinear resize (align_corners=False, no AA).
// One thread per output pixel. Exactly mirrors the reference:
//   src = max((o+0.5)*scale - 0.5, 0); i0=floor; i1=min(i0+1,in-1); w=src-i0
//   interpolate along y first, then x.
// ---------------------------------------------------------------------------
__global__ void gather_resize_kernel(const float* __restrict__ src,
                                     const int* __restrict__ sidx,
                                     float* __restrict__ dst,
                                     int C_src, int ch, int g, int chan_off,
                                     int C_out, int Hin, int Hout, float scale,
                                     int total) {
  int t = blockIdx.x * blockDim.x + threadIdx.x;
  if (t >= total) return;

  const int x = t % Hout;
  int tmp = t / Hout;
  const int y = tmp % Hout;
  tmp /= Hout;
  const int c = tmp % ch;
  const int b = tmp / ch;

  const int sc = sidx[b * C_src + ch * g + c];
  const float* sp = src + ((size_t)b * C_src + sc) * Hin * Hin;

  float sy = fmaxf((y + 0.5f) * scale - 0.5f, 0.0f);
  int y0 = (int)floorf(sy);
  if (y0 > Hin - 1) y0 = Hin - 1;
  const int y1 = min(y0 + 1, Hin - 1);
  const float wy = sy - (float)y0;

  float sx = fmaxf((x + 0.5f) * scale - 0.5f, 0.0f);
  int x0 = (int)floorf(sx);
  if (x0 > Hin - 1) x0 = Hin - 1;
  const int x1 = min(x0 + 1, Hin - 1);
  const float wx = sx - (float)x0;

  const float p00 = sp[y0 * Hin + x0];
  const float p10 = sp[y1 * Hin + x0];
  const float p01 = sp[y0 * Hin + x1];
  const float p11 = sp[y1 * Hin + x1];

  const float top = p00 * (1.0f - wy) + p10 * wy;   // column x0, interp in y
  const float bot = p01 * (1.0f - wy) + p11 * wy;   // column x1, interp in y
  const float v = top * (1.0f - wx) + bot * wx;     // interp in x

  dst[(((size_t)b * C_out + chan_off + c) * Hout + y) * Hout + x] = v;
}

// ---------------------------------------------------------------------------
// Host orchestration.
// ---------------------------------------------------------------------------
extern "C" void kernel_launch(void* const* d_in, const int* in_sizes, int n_in,
                              void* d_out, int out_size, void* d_ws, size_t ws_size,
                              hipStream_t stream) {
  (void)in_sizes; (void)n_in; (void)out_size; (void)ws_size;

  const float* lv[3] = {(const float*)d_in[0], (const float*)d_in[1],
                        (const float*)d_in[2]};
  const int B = 8;
  const int C[3] = {96, 192, 384};        // channels per level
  const int H[3] = {128, 64, 32};         // spatial size per level
  const int CH[3] = {32, 64, 128};        // group (= output slice) size per src
  const int CHOFF[3] = {0, 32, 96};       // channel offset of src j in output
  // AGS[j][i]: group of source level j feeding output level i
  const int AGS[3][3] = {{0, 1, 2}, {1, 0, 1}, {2, 2, 0}};

  // Workspace layout: means (5376 f32) then sorted indices (5376 i32)
  float* means = (float*)d_ws;
  int* sidx = (int*)((char*)d_ws + 5376 * sizeof(float));
  const int off[3] = {0, 768, 2304};      // per-level (b,c) flat offsets

  // Pass 1: GAP means via WMMA f32 row-sums (one wave per channel, 8 waves/block)
  for (int j = 0; j < 3; ++j) {
    const int waves = B * C[j];           // 768 / 1536 / 3072, all % 8 == 0
    gap_wmma_kernel<<<dim3(waves / 8), dim3(256), 0, stream>>>(
        lv[j], means + off[j], H[j] * H[j]);
  }

  // Pass 2: stable ascending argsort per (level, batch)
  for (int j = 0; j < 3; ++j) {
    rank_argsort_kernel<<<dim3(B), dim3(384), 0, stream>>>(
        means + off[j], sidx + off[j], C[j]);
  }

  // Pass 3: 9 gather+resize pieces. Output offsets (floats) in d_out.
  const size_t outOff[3] = {0, 29360128, 36700160};
  float* out = (float*)d_out;
  for (int i = 0; i < 3; ++i) {           // output level
    for (int j = 0; j < 3; ++j) {         // source level
      const int g = AGS[j][i];
      const long long total = (long long)B * CH[j] * H[i] * H[i];
      const int blocks = (int)((total + 255) / 256);
      const float scale = (float)H[j] / (float)H[i];
      gather_resize_kernel<<<dim3(blocks), dim3(256), 0, stream>>>(
          lv[j], sidx + off[j], out + outOff[i],
          C[j], CH[j], g, CHOFF[j], /*C_out=*/224, H[j], H[i], scale,
          (int)total);
    }
  }
}